// Attention_49366354100258
// MI455X (gfx1250) — compile-verified
//
#include <hip/hip_runtime.h>
#include <hip/hip_bf16.h>
#include <stdint.h>

// Problem constants (from reference)
#define B_SZ     64
#define T_SZ     2048
#define RNN_DIM  1024
#define EMB_DIM  512
#define ATT_DIM  128
#define N_FILT   32
#define KSIZE    31
#define PAD      15
#define NSPLIT   16           // T-splits for the context reduction

typedef __attribute__((ext_vector_type(2))) float v2f;
typedef __attribute__((ext_vector_type(8))) float v8f;

// ---------------------------------------------------------------------------
// Kernel 1: processed_query = ahs(64x1024) @ Wq^T(1024x128) via f32 WMMA.
// One wave per 16x16 output tile; K-loop of 256 x v_wmma_f32_16x16x4_f32.
// A-frag layout (ISA 7.12.2, 32-bit A 16x4): lane<16 -> K=0/1, lane>=16 -> K=2/3.
// ---------------------------------------------------------------------------
__global__ __launch_bounds__(32) void k1_query_gemm(const float* __restrict__ ahs,
                                                    const float* __restrict__ wq,
                                                    float* __restrict__ pq) {
  const int lane = threadIdx.x;
  const int at = blockIdx.x;          // 8 tiles over ATT_DIM
  const int mt = blockIdx.y;          // 4 tiles over B
  const int m = lane & 15;
  const int koff = (lane < 16) ? 0 : 2;
  const int half = lane >> 4;

  const float* arow = ahs + (size_t)(mt * 16 + m) * RNN_DIM; // A[m][k] = ahs[row][k]
  const float* brow = wq  + (size_t)(at * 16 + m) * RNN_DIM; // B[k][n] = wq[n][k]

  v8f acc = {};
  for (int kk = 0; kk < RNN_DIM / 4; ++kk) {
    v2f af = *(const v2f*)(arow + 4 * kk + koff);
    v2f bf = *(const v2f*)(brow + 4 * kk + koff);
    acc = __builtin_amdgcn_wmma_f32_16x16x4_f32(false, af, false, bf,
                                                (short)0, acc, false, false);
  }
#pragma unroll
  for (int r = 0; r < 8; ++r) {
    // C/D layout: VGPR r holds row r (lanes 0-15) / row r+8 (lanes 16-31), col = lane&15
    pq[(size_t)(mt * 16 + r + half * 8) * ATT_DIM + at * 16 + m] = acc[r];
  }
}

// ---------------------------------------------------------------------------
// Kernel 2: fused conv1d -> location dense (WMMA) -> tanh energy.
// One workgroup (8 waves) per (b, 128-wide t-block). Each wave owns 16 rows of t.
//   conv_s[wave][t][f]  (VALU conv, lane = filter)
//   loc(16x128) = conv(16x32) @ Wd^T(32x128): 8 N-chunks x 8 K-steps of WMMA f32
//   energy[t] = sum_a v[a]*tanh(loc + pq + processed_memory)   (pm read once)
// Energies written straight into the attention_weights slot of d_out.
// ---------------------------------------------------------------------------
__global__ __launch_bounds__(256) void k2_energy(const float* __restrict__ awc,
                                                 const float* __restrict__ wconv,
                                                 const float* __restrict__ wdense,
                                                 const float* __restrict__ pq,
                                                 const float* __restrict__ wv,
                                                 const float* __restrict__ pm,
                                                 float* __restrict__ energies) {
  __shared__ float x_s[2][160];          // input halo: 128 + 2*15 = 158 used
  __shared__ float wconv_s[N_FILT][2][KSIZE];
  __shared__ float dense_s[ATT_DIM][34]; // [a][f], padded stride (8B aligned rows)
  __shared__ float pq_s[ATT_DIM];
  __shared__ float v_s[ATT_DIM];
  __shared__ float conv_s[8][16][36];    // per-wave [t_local][f], padded stride

  const int tid = threadIdx.x;
  const int b = blockIdx.y;
  const int t_block = blockIdx.x * 128;

  // ---- stage inputs / weights into LDS ----
  for (int i = tid; i < 2 * 160; i += 256) {
    int c = i / 160, j = i % 160;
    float val = 0.f;
    int g = t_block - PAD + j;
    if (j < 158 && g >= 0 && g < T_SZ)
      val = awc[((size_t)b * 2 + c) * T_SZ + g];
    x_s[c][j] = val;
  }
  for (int i = tid; i < N_FILT * 2 * KSIZE; i += 256)
    (&wconv_s[0][0][0])[i] = wconv[i];
  for (int i = tid; i < ATT_DIM * N_FILT; i += 256)
    dense_s[i >> 5][i & 31] = wdense[i];
  if (tid < ATT_DIM) {
    pq_s[tid] = pq[(size_t)b * ATT_DIM + tid];
    v_s[tid] = wv[tid];
  }
  __syncthreads();

  const int wave = tid >> 5;
  const int lane = tid & 31;

  // ---- conv: lane = filter, loop over 16 local t rows ----
  {
    const int f = lane;
    for (int tl = 0; tl < 16; ++tl) {
      const int base = wave * 16 + tl;   // conv center offset into halo buffer
      float acc = 0.f;
#pragma unroll
      for (int c = 0; c < 2; ++c)
#pragma unroll
        for (int k = 0; k < KSIZE; ++k)
          acc = fmaf(wconv_s[f][c][k], x_s[c][base + k], acc);
      conv_s[wave][tl][f] = acc;
    }
  }
  __syncthreads();

  // ---- location dense via WMMA + fused energy ----
  const int m = lane & 15;
  const int koff = (lane < 16) ? 0 : 2;
  const int half = lane >> 4;

  float psum[8];
#pragma unroll
  for (int r = 0; r < 8; ++r) psum[r] = 0.f;

  const float* pm_base =
      pm + ((size_t)b * T_SZ + t_block + wave * 16) * ATT_DIM;

  for (int ac = 0; ac < 8; ++ac) {       // 8 chunks of 16 over ATT_DIM
    v8f acc = {};
#pragma unroll
    for (int kk = 0; kk < 8; ++kk) {     // K = 32 filters, 4 per step
      v2f af = *(const v2f*)(&conv_s[wave][m][4 * kk + koff]);
      v2f bf = *(const v2f*)(&dense_s[ac * 16 + m][4 * kk + koff]);
      acc = __builtin_amdgcn_wmma_f32_16x16x4_f32(false, af, false, bf,
                                                  (short)0, acc, false, false);
    }
    const int a = ac * 16 + m;
    const float pqv = pq_s[a];
    const float vv = v_s[a];
#pragma unroll
    for (int r = 0; r < 8; ++r) {
      const int trow = r + half * 8;     // local t row held by this VGPR/lane
      float val = acc[r] + pqv + pm_base[(size_t)trow * ATT_DIM + a];
      psum[r] += tanhf(val) * vv;
    }
  }

  // reduce over the 16 lanes of each half-wave (the a-dimension)
#pragma unroll
  for (int r = 0; r < 8; ++r) {
    float s = psum[r];
    s += __shfl_xor(s, 1, 32);
    s += __shfl_xor(s, 2, 32);
    s += __shfl_xor(s, 4, 32);
    s += __shfl_xor(s, 8, 32);
    psum[r] = s;
  }
  if (m == 0) {
#pragma unroll
    for (int r = 0; r < 8; ++r) {
      const int t = t_block + wave * 16 + r + half * 8;
      energies[(size_t)b * T_SZ + t] = psum[r];
    }
  }
}

// ---------------------------------------------------------------------------
// Kernel 3a: per-batch softmax over T, in place in the weights slot of d_out.
// ---------------------------------------------------------------------------
__global__ __launch_bounds__(256) void k3a_softmax(const uint8_t* __restrict__ mask,
                                                   float* __restrict__ weights) {
  __shared__ float w_s[T_SZ];
  __shared__ float red_s[8];

  const int b = blockIdx.x;
  const int tid = threadIdx.x;
  const int lane = tid & 31;
  const int wave = tid >> 5;

  float* wrow = weights + (size_t)b * T_SZ;
  const uint8_t* mrow = mask + (size_t)b * T_SZ;

  float lmax = -__builtin_inff();
#pragma unroll
  for (int i = 0; i < T_SZ / 256; ++i) {
    const int t = tid + i * 256;
    float e = wrow[t];
    if (mrow[t]) e = -__builtin_inff();
    w_s[t] = e;
    lmax = fmaxf(lmax, e);
  }
  for (int off = 16; off >= 1; off >>= 1) lmax = fmaxf(lmax, __shfl_xor(lmax, off, 32));
  if (lane == 0) red_s[wave] = lmax;
  __syncthreads();
  float bmax = red_s[0];
#pragma unroll
  for (int w = 1; w < 8; ++w) bmax = fmaxf(bmax, red_s[w]);
  __syncthreads();

  float lsum = 0.f;
#pragma unroll
  for (int i = 0; i < T_SZ / 256; ++i) {
    const int t = tid + i * 256;
    float p = __expf(w_s[t] - bmax);
    w_s[t] = p;
    lsum += p;
  }
  for (int off = 16; off >= 1; off >>= 1) lsum += __shfl_xor(lsum, off, 32);
  if (lane == 0) red_s[wave] = lsum;
  __syncthreads();
  float bsum = 0.f;
#pragma unroll
  for (int w = 0; w < 8; ++w) bsum += red_s[w];
  const float inv = 1.0f / bsum;

#pragma unroll
  for (int i = 0; i < T_SZ / 256; ++i) {
    const int t = tid + i * 256;
    wrow[t] = w_s[t] * inv;   // normalized attention weights -> d_out
  }
}

// ---------------------------------------------------------------------------
// Kernel 3b: partial context. Grid (NSPLIT, B): each WG owns 128 t-rows and
// streams memory with float4 (global_load_b128), 2 rows x 128 float4 / iter.
// Two row-parity accumulators are merged through LDS, partial written to ws.
// ---------------------------------------------------------------------------
__global__ __launch_bounds__(256) void k3b_ctx_partial(const float* __restrict__ memory,
                                                       const float* __restrict__ weights,
                                                       float* __restrict__ partial) {
  __shared__ float w_s[T_SZ / NSPLIT];       // 128 weights for this split
  __shared__ float4 red_s[EMB_DIM / 4];      // 128 float4 merge buffer

  const int split = blockIdx.x;
  const int b = blockIdx.y;
  const int tid = threadIdx.x;
  const int t0 = split * (T_SZ / NSPLIT);

  if (tid < T_SZ / NSPLIT)
    w_s[tid] = weights[(size_t)b * T_SZ + t0 + tid];
  __syncthreads();

  const int row = tid >> 7;                  // 0 or 1 (t parity within pair)
  const int col = tid & 127;                 // float4 column over E=512

  const float4* mem4 =
      (const float4*)(memory + ((size_t)b * T_SZ + t0) * EMB_DIM);

  float4 acc; acc.x = acc.y = acc.z = acc.w = 0.f;
  for (int it = 0; it < (T_SZ / NSPLIT) / 2; ++it) {
    const int tl = it * 2 + row;
    if ((it & 7) == 0 && it + 8 < (T_SZ / NSPLIT) / 2)
      __builtin_prefetch(&mem4[(size_t)(it * 2 + 16 + row) * (EMB_DIM / 4) + col], 0, 0);
    const float wv = w_s[tl];
    const float4 mv = mem4[(size_t)tl * (EMB_DIM / 4) + col];
    acc.x = fmaf(wv, mv.x, acc.x);
    acc.y = fmaf(wv, mv.y, acc.y);
    acc.z = fmaf(wv, mv.z, acc.z);
    acc.w = fmaf(wv, mv.w, acc.w);
  }

  // merge the two row-parity halves
  if (row == 1) red_s[col] = acc;
  __syncthreads();
  if (row == 0) {
    const float4 o = red_s[col];
    acc.x += o.x; acc.y += o.y; acc.z += o.z; acc.w += o.w;
    ((float4*)partial)[((size_t)b * NSPLIT + split) * (EMB_DIM / 4) + col] = acc;
  }
}

// ---------------------------------------------------------------------------
// Kernel 3c: deterministic combine of NSPLIT partials -> attention_context.
// ---------------------------------------------------------------------------
__global__ __launch_bounds__(256) void k3c_ctx_combine(const float* __restrict__ partial,
                                                       float* __restrict__ ctx) {
  const int b = blockIdx.x;
  const int tid = threadIdx.x;                // owns 2 consecutive E dims
  const float2* p2 = (const float2*)(partial + (size_t)b * NSPLIT * EMB_DIM);
  float2 acc; acc.x = 0.f; acc.y = 0.f;
#pragma unroll
  for (int s = 0; s < NSPLIT; ++s) {
    const float2 v = p2[(size_t)s * (EMB_DIM / 2) + tid];
    acc.x += v.x; acc.y += v.y;
  }
  ((float2*)ctx)[(size_t)b * (EMB_DIM / 2) + tid] = acc;
}

// ---------------------------------------------------------------------------
extern "C" void kernel_launch(void* const* d_in, const int* in_sizes, int n_in,
                              void* d_out, int out_size, void* d_ws, size_t ws_size,
                              hipStream_t stream) {
  const float*   ahs    = (const float*)d_in[0];   // (B, RNN)
  const float*   memory = (const float*)d_in[1];   // (B, T, E)
  const float*   pm     = (const float*)d_in[2];   // (B, T, A)
  const float*   awc    = (const float*)d_in[3];   // (B, 2, T)
  const uint8_t* mask   = (const uint8_t*)d_in[4]; // (B, T) bool
  const float*   wq     = (const float*)d_in[5];   // (A, RNN)
  const float*   wconv  = (const float*)d_in[6];   // (32, 2, 31)
  const float*   wdense = (const float*)d_in[7];   // (A, 32)
  const float*   wv     = (const float*)d_in[8];   // (1, A)

  float* ctx_out = (float*)d_out;                          // (B, E) first
  float* w_out   = (float*)d_out + (size_t)B_SZ * EMB_DIM; // (B, T) second

  float* pq_ws      = (float*)d_ws;                        // (B, A)        32 KB
  float* partial_ws = (float*)d_ws + (size_t)B_SZ * ATT_DIM; // (B,16,E)     2 MB

  // K1: processed_query (WMMA f32)
  k1_query_gemm<<<dim3(ATT_DIM / 16, B_SZ / 16), 32, 0, stream>>>(ahs, wq, pq_ws);

  // K2: conv + location dense (WMMA f32) + tanh energies -> weights slot
  k2_energy<<<dim3(T_SZ / 128, B_SZ), 256, 0, stream>>>(awc, wconv, wdense,
                                                        pq_ws, wv, pm, w_out);

  // K3a: softmax in place
  k3a_softmax<<<dim3(B_SZ), 256, 0, stream>>>(mask, w_out);

  // K3b: 1024-WG context partial sums (b128 streaming of `memory`)
  k3b_ctx_partial<<<dim3(NSPLIT, B_SZ), 256, 0, stream>>>(memory, w_out, partial_ws);

  // K3c: deterministic combine
  k3c_ctx_combine<<<dim3(B_SZ), 256, 0, stream>>>(partial_ws, ctx_out);
}